// CarafeUpsample_31430570672388
// MI455X (gfx1250) — compile-verified
//
#include <hip/hip_runtime.h>
#include <hip/hip_bf16.h>

// CARAFE upsample: compress(1x1) -> encoder(3x3) -> pixelshuffle+softmax -> 25-tap reassembly
// B=8, C=128, H=W=64, M=64, K=5, S=2, n_out=100, nH=nW=128.

#define B_   8
#define CIN  128
#define H_   64
#define W_   64
#define HW   4096       // 64*64
#define MCH  64
#define NOUT 100
#define KK   25
#define NH   128
#define NHW  16384      // 128*128

typedef __attribute__((ext_vector_type(2))) float v2f;
typedef __attribute__((ext_vector_type(8))) float v8f;

__device__ __forceinline__ v8f wmma_f32_4(v2f a, v2f b, v8f c) {
  // D = A(16x4 f32) * B(4x16 f32) + C(16x16 f32)
  return __builtin_amdgcn_wmma_f32_16x16x4_f32(
      /*neg_a=*/false, a, /*neg_b=*/false, b,
      /*c_mod=*/(short)0, c, /*reuse_a=*/false, /*reuse_b=*/false);
}

// ---------------------------------------------------------------------------
// Kernel 1: m[b,mo,hw] = sum_c Wc[mo,c]*x[b,c,hw] + bc[mo]
// GEMM [64 x 128] x [128 x 32768], one wave = 16x16 tile, K step 4.
// ---------------------------------------------------------------------------
__global__ __launch_bounds__(256) void k_compress(const float* __restrict__ x,
                                                  const float* __restrict__ wc,
                                                  const float* __restrict__ bc,
                                                  float* __restrict__ m) {
  const int lane = threadIdx.x & 31;
  const int wid  = threadIdx.x >> 5;
  const int gw   = blockIdx.x * 8 + wid;      // 8192 waves
  const int mt   = gw & 3;                    // 4 tiles over M=64
  const int nt   = gw >> 2;                   // 2048 tiles over pixels
  const int m0   = mt * 16;
  const int n0   = nt * 16;
  const int b    = n0 >> 12;                  // / HW
  const int hw0  = n0 & (HW - 1);
  const int col  = lane & 15;
  const int half = lane >> 4;

  // A lane view: row M = m0+col, K pair base = 2*half
  const float* __restrict__ arow = wc + (size_t)(m0 + col) * CIN + half * 2;
  // B lane view: pixel col = hw0+col, channel stride HW
  const float* __restrict__ bcol = x + (size_t)b * CIN * HW + hw0 + col;

  v8f acc = {};
#pragma unroll 8
  for (int k = 0; k < CIN; k += 4) {
    v2f a, bb;
    a.x = arow[k];
    a.y = arow[k + 1];
    const float* bp = bcol + (size_t)(k + half * 2) * HW;
    bb.x = bp[0];
    bb.y = bp[HW];
    acc = wmma_f32_4(a, bb, acc);
  }

  float* __restrict__ mp = m + ((size_t)b * MCH + m0) * HW + hw0 + col;
#pragma unroll
  for (int r = 0; r < 8; ++r) {
    const int mo = r + 8 * half;              // D row within tile
    mp[(size_t)mo * HW] = acc[r] + bc[m0 + mo];
  }
}

// ---------------------------------------------------------------------------
// Kernel 2: ker[b,o,hw] = bias[o] + sum_{mi,ky,kx} We[o,mi,ky,kx]*m[b,mi,y,x]
// im2col GEMM [112(pad) x 576] x [576 x 32768], K order = kidx*64 + mi.
// ---------------------------------------------------------------------------
__global__ __launch_bounds__(256) void k_encoder(const float* __restrict__ m,
                                                 const float* __restrict__ we,
                                                 const float* __restrict__ be,
                                                 float* __restrict__ ker) {
  const int lane = threadIdx.x & 31;
  const int wid  = threadIdx.x >> 5;
  const int gw   = blockIdx.x * 8 + wid;      // 14336 waves
  const int ot   = gw % 7;                    // 7 tiles cover O=100 (pad to 112)
  const int nt   = gw / 7;                    // 2048 pixel tiles
  const int o0   = ot * 16;
  const int n0   = nt * 16;
  const int b    = n0 >> 12;
  const int hw0  = n0 & (HW - 1);
  const int h    = hw0 >> 6;                  // all 16 pixels share row h
  const int col  = lane & 15;
  const int half = lane >> 4;
  const int w    = (hw0 & 63) + col;

  const int  orow = o0 + col;
  const bool aval = orow < NOUT;
  // A element for step (kidx, mi): we[orow, mi + 2*half + {0,1}, kidx]
  const float* __restrict__ ap = we + (size_t)orow * 576 + (size_t)(half * 2) * 9;
  const float* __restrict__ mb = m + (size_t)b * MCH * HW;

  v8f acc = {};
  for (int kidx = 0; kidx < 9; ++kidx) {
    const int dy = kidx / 3 - 1;
    const int dx = kidx % 3 - 1;
    const int y  = h + dy;
    const int xx = w + dx;
    const bool bval = ((unsigned)y < (unsigned)H_) && ((unsigned)xx < (unsigned)W_);
    const float* __restrict__ bp = mb + (size_t)(half * 2) * HW + y * W_ + xx;
#pragma unroll
    for (int mi = 0; mi < MCH; mi += 4) {
      v2f a, bb;
      a.x  = aval ? ap[mi * 9 + kidx] : 0.0f;
      a.y  = aval ? ap[(mi + 1) * 9 + kidx] : 0.0f;
      bb.x = bval ? bp[(size_t)mi * HW] : 0.0f;
      bb.y = bval ? bp[(size_t)(mi + 1) * HW] : 0.0f;
      acc = wmma_f32_4(a, bb, acc);
    }
  }

  float* __restrict__ kp = ker + (size_t)b * NOUT * HW + hw0 + col;
#pragma unroll
  for (int r = 0; r < 8; ++r) {
    const int o = o0 + r + 8 * half;
    if (o < NOUT) kp[(size_t)o * HW] = acc[r] + be[o];
  }
}

// ---------------------------------------------------------------------------
// Kernel 3: pixel-shuffle + softmax over the 25 kernel weights.
// wsm[b,kk,oh,ow] = softmax_kk( ker[b, kk*4 + (oh&1)*2 + (ow&1), oh/2, ow/2] )
// ---------------------------------------------------------------------------
__global__ __launch_bounds__(256) void k_softmax(const float* __restrict__ ker,
                                                 float* __restrict__ wsm) {
  const int idx = blockIdx.x * 256 + threadIdx.x;   // 131072 output pixels
  const int b  = idx >> 14;
  const int p  = idx & (NHW - 1);
  const int oh = p >> 7;
  const int ow = p & 127;
  const int h  = oh >> 1;
  const int w  = ow >> 1;
  const int g  = (oh & 1) * 2 + (ow & 1);

  const float* __restrict__ kp = ker + (size_t)b * NOUT * HW + (size_t)g * HW + h * W_ + w;
  float v[KK];
  float mx = -1e30f;
#pragma unroll
  for (int kk = 0; kk < KK; ++kk) {
    v[kk] = kp[(size_t)(kk * 4) * HW];
    mx = fmaxf(mx, v[kk]);
  }
  float sum = 0.0f;
#pragma unroll
  for (int kk = 0; kk < KK; ++kk) {
    v[kk] = __expf(v[kk] - mx);
    sum += v[kk];
  }
  const float inv = __frcp_rn(sum);
  float* __restrict__ op = wsm + (size_t)b * KK * NHW + p;
#pragma unroll
  for (int kk = 0; kk < KK; ++kk) op[(size_t)kk * NHW] = v[kk] * inv;
}

// ---------------------------------------------------------------------------
// Kernel 4: out[b,c,oh,ow] = sum_kk wsm[b,kk,oh,ow] * xpad[b,c, oh/2+ky-2, ow/2+kx-2]
// One block per (b, oh): weights row -> LDS; 256 thr = 128 ow x 2 channel groups.
// ---------------------------------------------------------------------------
__global__ __launch_bounds__(256) void k_reassemble(const float* __restrict__ x,
                                                    const float* __restrict__ wsm,
                                                    float* __restrict__ out) {
  __shared__ float lw[KK * NH];               // 25*128 f32 = 12.8 KB
  const int blk = blockIdx.x;                 // B*nH = 1024
  const int b   = blk >> 7;
  const int oh  = blk & 127;
  const int tid = threadIdx.x;

  const float* __restrict__ wp = wsm + (size_t)b * KK * NHW + (size_t)oh * NH;
  for (int i = tid; i < KK * NH; i += 256) {
    const int kk = i >> 7;
    const int ow = i & 127;
    lw[i] = wp[(size_t)kk * NHW + ow];
  }
  __syncthreads();

  const int ow = tid & 127;
  const int cg = tid >> 7;                    // 0..1
  const int h  = oh >> 1;
  const int w  = ow >> 1;
  const float* __restrict__ xb = x + (size_t)b * CIN * HW;
  float* __restrict__ ob = out + (size_t)b * CIN * NHW + (size_t)oh * NH + ow;

  for (int c = cg * 64; c < cg * 64 + 64; ++c) {
    const float* __restrict__ xc = xb + (size_t)c * HW;
    float acc = 0.0f;
#pragma unroll
    for (int ky = 0; ky < 5; ++ky) {
      const int y = h + ky - 2;
      if ((unsigned)y < (unsigned)H_) {
        const float* __restrict__ xr = xc + y * W_;
#pragma unroll
        for (int kx = 0; kx < 5; ++kx) {
          const int xcol = w + kx - 2;
          const float val = ((unsigned)xcol < (unsigned)W_) ? xr[xcol] : 0.0f;
          acc = fmaf(lw[(ky * 5 + kx) * NH + ow], val, acc);
        }
      }
    }
    ob[(size_t)c * NHW] = acc;
  }
}

// ---------------------------------------------------------------------------
// Workspace layout (fp32 elements):
//   [0, 3276800)          ker_raw  (B*100*HW)                      13.1 MB
//   [3276800, 6553600)    m (B*64*HW, first 2097152)  then reused
//                         as wsm (B*25*nHW, full 3276800)          13.1 MB
// total required ws_size >= 26,214,400 bytes.
// ---------------------------------------------------------------------------
extern "C" void kernel_launch(void* const* d_in, const int* in_sizes, int n_in,
                              void* d_out, int out_size, void* d_ws, size_t ws_size,
                              hipStream_t stream) {
  const float* x  = (const float*)d_in[0];
  const float* cw = (const float*)d_in[1];
  const float* cb = (const float*)d_in[2];
  const float* ew = (const float*)d_in[3];
  const float* eb = (const float*)d_in[4];
  float* out = (float*)d_out;

  float* ker  = (float*)d_ws;                 // 3,276,800 floats
  float* mbuf = ker + (size_t)B_ * NOUT * HW; // m, later overwritten by wsm
  float* wsm  = mbuf;

  // 1) 1x1 compress conv (WMMA f32 GEMM): 8192 waves / 8 per block
  k_compress<<<1024, 256, 0, stream>>>(x, cw, cb, mbuf);
  // 2) 3x3 encoder conv (WMMA f32 GEMM + im2col): 14336 waves / 8 per block
  k_encoder<<<1792, 256, 0, stream>>>(mbuf, ew, eb, ker);
  // 3) pixel-shuffle + softmax(25)
  k_softmax<<<512, 256, 0, stream>>>(ker, wsm);
  // 4) weighted 25-tap reassembly
  k_reassemble<<<1024, 256, 0, stream>>>(x, wsm, out);
}